// Adaptive_EDDG_4870492913703
// MI455X (gfx1250) — compile-verified
//
#include <hip/hip_runtime.h>
#include <hip/hip_bf16.h>

// ---------------------------------------------------------------------------
// Adaptive_EDDG for MI455X (gfx1250), fp32 end-to-end.
// All 1x1 convs run as wave-level GEMMs on V_WMMA_F32_16X16X4_F32.
// ED branch stages the whole 48KB point cloud in LDS (320KB/WGP available).
// ---------------------------------------------------------------------------

#define BB 4
#define NN 4096
#define CZ 164   // concat channels: 128 + 32 + 4

typedef __attribute__((ext_vector_type(2))) float v2f;
typedef __attribute__((ext_vector_type(8))) float v8f;

__device__ inline v8f v8f_zero() {
  v8f z;
#pragma unroll
  for (int r = 0; r < 8; ++r) z[r] = 0.f;
  return z;
}

// ---------------------------------------------------------------------------
// Y[b,o,n] = relu( scale[o] * (sum_c W[o,c] * X[b,c,n] + bias[o]) + shift[o] )
// X: [BB, C, NN] row-major (n contiguous), W: [O, C], Y: [BB, O, NN].
// Block = 256 threads = 8 waves; wave w handles n-tile (blockIdx.x*8+w)*16,
// and MT stacked 16-row o-tiles (o0 = blockIdx.y*16*MT), K-step of 4.
//
// A frag (16x4 f32, ISA layout): lanes 0-15 hold K={0,1} in VGPR{0,1},
//   lanes 16-31 hold K={2,3}.  B frag mirrors the C/D half-lane split:
//   VGPR0 = rows K0 (lanes 0-15) / K2 (lanes 16-31), VGPR1 = K1 / K3.
// C/D frag: VGPR r -> M = r + 8*(lane>=16), N = lane&15.
// ---------------------------------------------------------------------------
template<int MT>
__global__ __launch_bounds__(256) void gemm_bn_relu_kernel(
    const float* __restrict__ X, const float* __restrict__ W,
    const float* __restrict__ bias, const float* __restrict__ scale,
    const float* __restrict__ shift, float* __restrict__ Y,
    int C, int O)
{
  const int lane = threadIdx.x & 31;
  const int wave = threadIdx.x >> 5;
  const int n0   = (blockIdx.x * 8 + wave) * 16;
  const int o0   = blockIdx.y * (16 * MT);
  const int b    = blockIdx.z;

  const float* __restrict__ Xb = X + (size_t)b * C * NN;
  float* __restrict__ Yb = Y + (size_t)b * O * NN;

  const int mrow  = lane & 15;
  const int khalf = (lane >> 4) << 1;   // 0 for lanes 0-15, 2 for lanes 16-31

  v8f acc[MT];
#pragma unroll
  for (int m = 0; m < MT; ++m) acc[m] = v8f_zero();

  for (int k = 0; k < C; k += 4) {
    // B fragment: X[k+khalf + {0,1}][n0 + mrow]  (coalesced 64B per half-wave)
    const float* xcol = Xb + (size_t)(k + khalf) * NN + (n0 + mrow);
    v2f bf;
    bf[0] = xcol[0];
    bf[1] = xcol[NN];
#pragma unroll
    for (int m = 0; m < MT; ++m) {
      const float* wrow = W + (size_t)(o0 + m * 16 + mrow) * C + (k + khalf);
      v2f af;
      af[0] = wrow[0];
      af[1] = wrow[1];
      acc[m] = __builtin_amdgcn_wmma_f32_16x16x4_f32(
          /*neg_a=*/false, af, /*neg_b=*/false, bf,
          /*c_mod=*/(short)0, acc[m], /*reuse_a=*/false, /*reuse_b=*/false);
    }
  }

  const int ncol  = lane & 15;
  const int mbase = (lane >> 4) << 3;   // 0 or 8
#pragma unroll
  for (int m = 0; m < MT; ++m) {
#pragma unroll
    for (int r = 0; r < 8; ++r) {
      const int o = o0 + m * 16 + mbase + r;
      float v = acc[m][r] + bias[o];
      v = fmaf(v, scale[o], shift[o]);
      Yb[(size_t)o * NN + n0 + ncol] = fmaxf(v, 0.f);
    }
  }
}

// ---------------------------------------------------------------------------
// Pass 1: per-batch max clamped pairwise d^2 (reference formula x2i+x2j-2dot).
// Whole cloud staged in LDS (48 KB). atomicMax on bit pattern is exact for
// non-negative floats.
// ---------------------------------------------------------------------------
__global__ __launch_bounds__(256) void maxdist_kernel(
    const float* __restrict__ xyz, float* __restrict__ d2max)
{
  __shared__ float sxv[NN], syv[NN], szv[NN];
  const int b = blockIdx.y;
  const float* p = xyz + (size_t)b * NN * 3;
  for (int j = threadIdx.x; j < NN; j += blockDim.x) {
    sxv[j] = p[j * 3 + 0]; syv[j] = p[j * 3 + 1]; szv[j] = p[j * 3 + 2];
  }
  __syncthreads();

  const int i = blockIdx.x * blockDim.x + threadIdx.x;
  const float xi = sxv[i], yi = syv[i], zi = szv[i];
  const float x2i = xi * xi + yi * yi + zi * zi;
  float m = 0.f;
  for (int j = 0; j < NN; ++j) {
    const float xj = sxv[j], yj = syv[j], zj = szv[j];
    const float x2j = xj * xj + yj * yj + zj * zj;
    const float d2 = x2i + x2j - 2.f * (xi * xj + yi * yj + zi * zj);
    m = fmaxf(m, d2);
  }
  atomicMax((unsigned int*)(d2max + b), __float_as_uint(fmaxf(m, 0.f)));
}

// ---------------------------------------------------------------------------
// Pass 2: masked neighbor covariance, closed-form 3x3 eigvalsh (ascending),
// tiny 3->4->4 MLP; writes channels 160..163 of the concat buffer Z.
// ---------------------------------------------------------------------------
__global__ __launch_bounds__(256) void eigen_mlp_kernel(
    const float* __restrict__ xyz, const float* __restrict__ d2max,
    const float* __restrict__ ew1, const float* __restrict__ eb1,
    const float* __restrict__ ew2, const float* __restrict__ eb2,
    float* __restrict__ Z)
{
  __shared__ float sxv[NN], syv[NN], szv[NN];
  const int b = blockIdx.y;
  const float* p = xyz + (size_t)b * NN * 3;
  for (int j = threadIdx.x; j < NN; j += blockDim.x) {
    sxv[j] = p[j * 3 + 0]; syv[j] = p[j * 3 + 1]; szv[j] = p[j * 3 + 2];
  }
  __syncthreads();

  const float radius = sqrtf(d2max[b]) * 0.1f;
  const int i = blockIdx.x * blockDim.x + threadIdx.x;
  const float xi = sxv[i], yi = syv[i], zi = szv[i];
  const float x2i = xi * xi + yi * yi + zi * zi;

  float cnt = 0.f, s1x = 0.f, s1y = 0.f, s1z = 0.f;
  float sxx = 0.f, sxy = 0.f, sxz = 0.f, syy = 0.f, syz = 0.f, szz = 0.f;
  for (int j = 0; j < NN; ++j) {
    const float xj = sxv[j], yj = syv[j], zj = szv[j];
    const float x2j = xj * xj + yj * yj + zj * zj;
    const float d2 = x2i + x2j - 2.f * (xi * xj + yi * yj + zi * zj);
    const float d  = sqrtf(fmaxf(d2, 0.f));
    const float w  = ((d < radius) && (j != i)) ? 1.f : 0.f;
    cnt += w;
    s1x = fmaf(w, xj, s1x); s1y = fmaf(w, yj, s1y); s1z = fmaf(w, zj, s1z);
    sxx = fmaf(w, xj * xj, sxx); sxy = fmaf(w, xj * yj, sxy);
    sxz = fmaf(w, xj * zj, sxz); syy = fmaf(w, yj * yj, syy);
    syz = fmaf(w, yj * zj, syz); szz = fmaf(w, zj * zj, szz);
  }
  const float ic = 1.f / fmaxf(cnt, 1.f);
  const float mx = s1x * ic, my = s1y * ic, mz = s1z * ic;
  const float invN = 1.f / (float)NN;
  const float a00 = (sxx - cnt * mx * mx) * invN;
  const float a01 = (sxy - cnt * mx * my) * invN;
  const float a02 = (sxz - cnt * mx * mz) * invN;
  const float a11 = (syy - cnt * my * my) * invN;
  const float a12 = (syz - cnt * my * mz) * invN;
  const float a22 = (szz - cnt * mz * mz) * invN;

  // symmetric 3x3 eigvalsh (trigonometric), ascending order
  float e0, e1, e2;
  {
    const float q = (a00 + a11 + a22) * (1.f / 3.f);
    const float b00 = a00 - q, b11 = a11 - q, b22 = a22 - q;
    const float p2 = b00 * b00 + b11 * b11 + b22 * b22 +
                     2.f * (a01 * a01 + a02 * a02 + a12 * a12);
    if (p2 <= 1e-30f) {
      e0 = e1 = e2 = q;
    } else {
      const float pp = sqrtf(p2 * (1.f / 6.f));
      const float inv = 1.f / pp;
      const float c00 = b00 * inv, c11 = b11 * inv, c22 = b22 * inv;
      const float c01 = a01 * inv, c02 = a02 * inv, c12 = a12 * inv;
      float detB = c00 * (c11 * c22 - c12 * c12)
                 - c01 * (c01 * c22 - c12 * c02)
                 + c02 * (c01 * c12 - c11 * c02);
      float r = fminf(1.f, fmaxf(-1.f, detB * 0.5f));
      const float phi = acosf(r) * (1.f / 3.f);
      e2 = q + 2.f * pp * cosf(phi);
      e0 = q + 2.f * pp * cosf(phi + 2.0943951023931953f); // + 2*pi/3
      e1 = 3.f * q - e0 - e2;
    }
  }

  // MLP: h = relu(eig @ ew1^T + eb1); out = h @ ew2^T + eb2
  float h[4];
#pragma unroll
  for (int o = 0; o < 4; ++o)
    h[o] = fmaxf(fmaf(e0, ew1[o * 3 + 0],
                 fmaf(e1, ew1[o * 3 + 1],
                 fmaf(e2, ew1[o * 3 + 2], eb1[o]))), 0.f);
#pragma unroll
  for (int o = 0; o < 4; ++o) {
    const float v = fmaf(h[0], ew2[o * 4 + 0],
                    fmaf(h[1], ew2[o * 4 + 1],
                    fmaf(h[2], ew2[o * 4 + 2],
                    fmaf(h[3], ew2[o * 4 + 3], eb2[o]))));
    Z[((size_t)b * CZ + 160 + o) * NN + i] = v;
  }
}

// src [BB, Csrc, NN] -> Z channels [coff, coff+Csrc)
__global__ void copy_into_Z_kernel(const float* __restrict__ src,
                                   float* __restrict__ Z, int Csrc, int coff)
{
  const size_t idx = (size_t)blockIdx.x * blockDim.x + threadIdx.x;
  const size_t total = (size_t)BB * Csrc * NN;
  if (idx >= total) return;
  const int n = (int)(idx % NN);
  const size_t t = idx / NN;
  const int c = (int)(t % Csrc);
  const int b = (int)(t / Csrc);
  Z[((size_t)b * CZ + coff + c) * NN + n] = src[idx];
}

__global__ void copy_f32_kernel(const float* __restrict__ src,
                                float* __restrict__ dst, int n)
{
  const int idx = blockIdx.x * blockDim.x + threadIdx.x;
  if (idx < n) dst[idx] = src[idx];
}

__global__ void zero_kernel(float* p, int n)
{
  const int i = threadIdx.x;
  if (i < n) p[i] = 0.f;
}

// ---------------------------------------------------------------------------
extern "C" void kernel_launch(void* const* d_in, const int* in_sizes, int n_in,
                              void* d_out, int out_size, void* d_ws, size_t ws_size,
                              hipStream_t stream)
{
  (void)in_sizes; (void)n_in; (void)out_size; (void)ws_size;

  const float* xyz   = (const float*)d_in[0];
  const float* h1    = (const float*)d_in[1];
  const float* h2_in = (const float*)d_in[2];
  const float* dg_w1 = (const float*)d_in[3];
  const float* dg_b1 = (const float*)d_in[4];
  const float* dg_s1 = (const float*)d_in[5];
  const float* dg_t1 = (const float*)d_in[6];
  const float* dg_w2 = (const float*)d_in[7];
  const float* dg_b2 = (const float*)d_in[8];
  const float* dg_s2 = (const float*)d_in[9];
  const float* dg_t2 = (const float*)d_in[10];
  const float* dg_w3 = (const float*)d_in[11];
  const float* dg_b3 = (const float*)d_in[12];
  const float* dg_s3 = (const float*)d_in[13];
  const float* dg_t3 = (const float*)d_in[14];
  const float* ed_w1 = (const float*)d_in[15];
  const float* ed_b1 = (const float*)d_in[16];
  const float* ed_w2 = (const float*)d_in[17];
  const float* ed_b2 = (const float*)d_in[18];
  const float* w1    = (const float*)d_in[19];
  const float* b1    = (const float*)d_in[20];
  const float* s1    = (const float*)d_in[21];
  const float* t1    = (const float*)d_in[22];
  const float* w2    = (const float*)d_in[23];
  const float* b2    = (const float*)d_in[24];
  const float* s2    = (const float*)d_in[25];
  const float* t2    = (const float*)d_in[26];
  const float* w3    = (const float*)d_in[27];
  const float* b3    = (const float*)d_in[28];
  const float* s3    = (const float*)d_in[29];
  const float* t3    = (const float*)d_in[30];

  float* out   = (float*)d_out;
  float* z_out = out + (size_t)BB * NN * 3;   // z: [BB, 128, NN]

  // workspace layout (with reuse): peak ~61 MB
  char* ws = (char*)d_ws;
  float* d2max = (float*)ws;                                   // BB floats
  float* bufA  = (float*)(ws + 256);                           // 32 MB: h2a then f1
  float* bufB  = (float*)(ws + 256 + ((size_t)32 << 20));      // 16 MB: h2b then f2
  float* h2c   = (float*)(ws + 256 + ((size_t)48 << 20));      //  2 MB
  float* Zbuf  = (float*)(ws + 256 + ((size_t)50 << 20));      // ~10.3 MB

  // ED branch pass 1: global max pairwise distance per batch
  zero_kernel<<<1, 32, 0, stream>>>(d2max, BB);
  maxdist_kernel<<<dim3(NN / 256, BB), 256, 0, stream>>>(xyz, d2max);

  // DG branch: 1024 -> 256 -> 64 -> 32
  gemm_bn_relu_kernel<4><<<dim3(NN / 128, 256 / 64, BB), 256, 0, stream>>>(
      h2_in, dg_w1, dg_b1, dg_s1, dg_t1, bufA, 1024, 256);
  gemm_bn_relu_kernel<4><<<dim3(NN / 128, 64 / 64, BB), 256, 0, stream>>>(
      bufA, dg_w2, dg_b2, dg_s2, dg_t2, bufB, 256, 64);
  gemm_bn_relu_kernel<2><<<dim3(NN / 128, 32 / 32, BB), 256, 0, stream>>>(
      bufB, dg_w3, dg_b3, dg_s3, dg_t3, h2c, 64, 32);

  // concat: [h1 | h2 | h3] -> Z [BB, 164, NN]
  {
    const size_t tot = (size_t)BB * 128 * NN;
    copy_into_Z_kernel<<<(unsigned)((tot + 255) / 256), 256, 0, stream>>>(h1, Zbuf, 128, 0);
  }
  {
    const size_t tot = (size_t)BB * 32 * NN;
    copy_into_Z_kernel<<<(unsigned)((tot + 255) / 256), 256, 0, stream>>>(h2c, Zbuf, 32, 128);
  }
  eigen_mlp_kernel<<<dim3(NN / 256, BB), 256, 0, stream>>>(
      xyz, d2max, ed_w1, ed_b1, ed_w2, ed_b2, Zbuf);

  // head: 164 -> 512 -> 256 -> 128
  gemm_bn_relu_kernel<4><<<dim3(NN / 128, 512 / 64, BB), 256, 0, stream>>>(
      Zbuf, w1, b1, s1, t1, bufA, CZ, 512);
  gemm_bn_relu_kernel<4><<<dim3(NN / 128, 256 / 64, BB), 256, 0, stream>>>(
      bufA, w2, b2, s2, t2, bufB, 512, 256);
  gemm_bn_relu_kernel<4><<<dim3(NN / 128, 128 / 64, BB), 256, 0, stream>>>(
      bufB, w3, b3, s3, t3, z_out, 256, 128);

  // xyz passthrough (first output of the tuple)
  {
    const int tot = BB * NN * 3;
    copy_f32_kernel<<<(tot + 255) / 256, 256, 0, stream>>>(xyz, out, tot);
  }
}